// SentenceRewardModel_60249801228442
// MI455X (gfx1250) — compile-verified
//
#include <hip/hip_runtime.h>
#include <math.h>

#define B 8
#define S 4096
#define H 4096
#define D 128
#define NMAX 64
#define THETA 10000.0f

typedef __attribute__((ext_vector_type(2))) float v2f;
typedef __attribute__((ext_vector_type(8))) float v8f;

// ---------------------------------------------------------------------------
// Kernel 1: build sent_pos (gather positions of mask==1, rank-ordered) +count
// ---------------------------------------------------------------------------
__global__ void __launch_bounds__(256) k_positions(const int* __restrict__ mask,
                                                   int* __restrict__ sent_pos,
                                                   int* __restrict__ count)
{
    const int b = blockIdx.x;
    const int t = threadIdx.x;              // 0..255, each owns 16 positions
    __shared__ int sc[256];

    const int base = b * S + t * 16;
    int local[16];
    int cnt = 0;
#pragma unroll
    for (int i = 0; i < 16; ++i) {
        int m = mask[base + i];
        local[i] = m;
        cnt += (m > 0) ? 1 : 0;
    }
    sc[t] = cnt;
    __syncthreads();
    // inclusive Hillis-Steele scan over 256 entries
    for (int off = 1; off < 256; off <<= 1) {
        int v = (t >= off) ? sc[t - off] : 0;
        __syncthreads();
        sc[t] += v;
        __syncthreads();
    }
    int rank = sc[t] - cnt;                 // exclusive prefix
    if (t < NMAX) sent_pos[b * NMAX + t] = 0;   // match jnp.zeros init
    __syncthreads();
#pragma unroll
    for (int i = 0; i < 16; ++i) {
        if (local[i] > 0) {
            if (rank < NMAX) sent_pos[b * NMAX + rank] = t * 16 + i;
            ++rank;
        }
    }
    if (t == 255) count[b] = sc[255];
}

// ---------------------------------------------------------------------------
// Kernel 2: ks = hidden[gathered rows] @ Wk + bk  via V_WMMA_F32_16X16X4_F32
//   A tile staged with GLOBAL_LOAD_ASYNC_TO_LDS_B128 (ASYNCcnt-tracked)
//   B tile staged K-pair-interleaved so B fragments are single b64 LDS loads
// ---------------------------------------------------------------------------
#define APITCH 68    // 64 + 4 pad (floats), row pitch of A tile
#define BP     288   // dwords per K-pair row of B tile: 256 data + 32 pad

__global__ void __launch_bounds__(256) k_kproj_wmma(const float* __restrict__ hidden,
                                                    const float* __restrict__ Wk,
                                                    const float* __restrict__ bk,
                                                    const int*   __restrict__ sent_pos,
                                                    float* __restrict__ ks)
{
    const int b    = blockIdx.x;
    const int tid  = threadIdx.x;
    const int lane = tid & 31;
    const int wave = tid >> 5;              // 0..7
    const int mt   = wave & 3;              // M tile 0..3 (rows mt*16..)
    const int nh   = wave >> 2;             // 0..1 -> N tiles nh*4 .. nh*4+3

    __shared__ int   pos[NMAX];
    __shared__ float As[NMAX * APITCH];     // 64 x 64 fp32 gathered rows (17.4 KB)
    __shared__ float Bs[32 * BP];           // 32 K-pair rows x 256 interleaved (36.9 KB)

    if (tid < NMAX) pos[tid] = sent_pos[b * NMAX + tid];
    __syncthreads();

    v8f acc[4] = {};
    const float* hb = hidden + (size_t)b * S * H;

    for (int kc = 0; kc < H; kc += 64) {
        // ---- stage A asynchronously: 64 rows x 64 cols = 1024 b128 slots ----
#pragma unroll
        for (int i = 0; i < 4; ++i) {
            int f   = tid + 256 * i;        // b128 slot 0..1023
            int row = f >> 4;
            int c4  = (f & 15) << 2;
            const float* gp = hb + (size_t)pos[row] * H + kc + c4;
            unsigned ldsa = (unsigned)(uintptr_t)(&As[row * APITCH + c4]);
            asm volatile("global_load_async_to_lds_b128 %0, %1, off"
                         :: "v"(ldsa), "v"(gp) : "memory");
        }
        // ---- stage B (K-pair interleaved): Bs[kp][2n+(k&1)] = Wk[kc+k][n] ----
#pragma unroll
        for (int i = 0; i < 8; ++i) {
            int f  = tid + 256 * i;         // 0..2047
            int kp = f >> 6;                // K pair row 0..31
            int n  = (f & 63) << 1;         // even column 0..126
            float2 a0 = *(const float2*)(Wk + (size_t)(kc + 2 * kp)     * D + n);
            float2 a1 = *(const float2*)(Wk + (size_t)(kc + 2 * kp + 1) * D + n);
            float4 vv = make_float4(a0.x, a1.x, a0.y, a1.y);
            *(float4*)&Bs[kp * BP + 2 * n] = vv;
        }
        asm volatile("s_wait_asynccnt 0x0" ::: "memory");
        __syncthreads();

        // ---- WMMA over this 64-wide K chunk ----
        const int mrow = lane & 15;
        const int khal = lane >> 4;         // lanes 0-15 -> K+0/1, 16-31 -> K+2/3
#pragma unroll
        for (int kk = 0; kk < 64; kk += 4) {
            v2f a = *(const v2f*)&As[(mt * 16 + mrow) * APITCH + kk + 2 * khal];
            const int kp = (kk >> 1) + khal;
#pragma unroll
            for (int j = 0; j < 4; ++j) {
                const int n = (nh * 4 + j) * 16 + mrow;
                v2f bb = *(const v2f*)&Bs[kp * BP + 2 * n];
                acc[j] = __builtin_amdgcn_wmma_f32_16x16x4_f32(
                    false, a, false, bb, (short)0, acc[j], false, false);
            }
        }
        __syncthreads();
    }

    // write D tiles (+bias). D layout: VGPR r -> row r (lanes 0-15) / r+8 (16-31)
    const int n0  = lane & 15;
    const int mhi = (lane >> 4) * 8;
#pragma unroll
    for (int j = 0; j < 4; ++j) {
        const int ncol = (nh * 4 + j) * 16 + n0;
        const float bias = bk[ncol];
#pragma unroll
        for (int r = 0; r < 8; ++r) {
            const int m = mt * 16 + r + mhi;
            ks[((size_t)b * NMAX + m) * D + ncol] = acc[j][r] + bias;
        }
    }
}

// ---------------------------------------------------------------------------
// Kernel 3: q = hidden[b, S-1] @ Wq + bq ;  vs[n] = hidden[b,pos[n]] . Wr + br
// ---------------------------------------------------------------------------
__global__ void __launch_bounds__(256) k_small_proj(const float* __restrict__ hidden,
                                                    const float* __restrict__ Wq,
                                                    const float* __restrict__ bq,
                                                    const float* __restrict__ Wr,
                                                    const float* __restrict__ br,
                                                    const int*   __restrict__ sent_pos,
                                                    float* __restrict__ qv,
                                                    float* __restrict__ vs)
{
    const int b    = blockIdx.x;
    const int tid  = threadIdx.x;
    const int lane = tid & 31;
    const int wave = tid >> 5;

    __shared__ float hlast[H];              // 16 KB
    __shared__ float wr[H];                 // 16 KB
    const float* hb = hidden + (size_t)b * S * H;
    const float* hl = hb + (size_t)(S - 1) * H;
    for (int k = tid; k < H; k += 256) { hlast[k] = hl[k]; wr[k] = Wr[k]; }
    __syncthreads();

    // q: one output column per thread (tid < 128); Wq reads coalesce over tid
    if (tid < D) {
        float s = 0.f;
        for (int k = 0; k < H; ++k) s += hlast[k] * Wq[(size_t)k * D + tid];
        qv[b * D + tid] = s + bq[tid];
    }

    // vs: each wave handles 8 gathered rows; lane-strided dot + wave reduce
    for (int r8 = 0; r8 < 8; ++r8) {
        const int r = wave * 8 + r8;
        const float* hrow = hb + (size_t)sent_pos[b * NMAX + r] * H;
        float s = 0.f;
        for (int k = lane; k < H; k += 32) s += hrow[k] * wr[k];
#pragma unroll
        for (int off = 16; off > 0; off >>= 1) s += __shfl_xor(s, off, 32);
        if (lane == 0) vs[b * NMAX + r] = s + br[0];
    }
}

// ---------------------------------------------------------------------------
// Kernel 4: RoPE + masked softmax attention + rewards + pairwise loss
//   one wave32 per batch (8 waves); lane owns rope pairs p = lane, lane+32
// ---------------------------------------------------------------------------
__global__ void __launch_bounds__(256) k_finalize(const float* __restrict__ qv,
                                                  const float* __restrict__ ks,
                                                  const float* __restrict__ vs,
                                                  const int*   __restrict__ count,
                                                  float* __restrict__ out)
{
    const int tid  = threadIdx.x;
    const int lane = tid & 31;
    const int b    = tid >> 5;              // wave per batch

    __shared__ float logits[B][NMAX];
    __shared__ float rew[B];

    const int c = count[b];

    // rope'd query (rotation index = count-1)
    float rqa[2], rqb[2], freq[2];
#pragma unroll
    for (int u = 0; u < 2; ++u) {
        const int p = lane + 32 * u;        // rope pair 0..63
        freq[u] = powf(THETA, -(float)p / 64.f);
        const float ang = (float)(c - 1) * freq[u];
        float sn, cs; sincosf(ang, &sn, &cs);
        const float qa = qv[b * D + 2 * p];
        const float qb = qv[b * D + 2 * p + 1];
        rqa[u] = qa * cs - qb * sn;
        rqb[u] = qa * sn + qb * cs;
    }

    const float inv_sqrt_d = rsqrtf((float)D);
    for (int n = 0; n < c; ++n) {
        float part = 0.f;
#pragma unroll
        for (int u = 0; u < 2; ++u) {
            const int p = lane + 32 * u;
            const float ang = (float)n * freq[u];
            float sn, cs; sincosf(ang, &sn, &cs);
            const float ka = ks[((size_t)b * NMAX + n) * D + 2 * p];
            const float kb = ks[((size_t)b * NMAX + n) * D + 2 * p + 1];
            const float rka = ka * cs - kb * sn;
            const float rkb = ka * sn + kb * cs;
            part += rqa[u] * rka + rqb[u] * rkb;
        }
#pragma unroll
        for (int off = 16; off > 0; off >>= 1) part += __shfl_xor(part, off, 32);
        if (lane == 0) logits[b][n] = part * inv_sqrt_d;
    }
    __syncthreads();

    // masked softmax over n < c, reward = sum attn * (vs[n]-vs[n-1])
    const float NEG = -__builtin_inff();
    float l0 = (lane      < c) ? logits[b][lane]      : NEG;
    float l1 = (lane + 32 < c) ? logits[b][lane + 32] : NEG;
    float mx = fmaxf(l0, l1);
#pragma unroll
    for (int off = 16; off > 0; off >>= 1) mx = fmaxf(mx, __shfl_xor(mx, off, 32));
    float e0 = (lane      < c) ? expf(l0 - mx) : 0.f;
    float e1 = (lane + 32 < c) ? expf(l1 - mx) : 0.f;
    float se = e0 + e1;
#pragma unroll
    for (int off = 16; off > 0; off >>= 1) se += __shfl_xor(se, off, 32);
    float sr0 = (lane < c) ? (vs[b * NMAX + lane] - (lane ? vs[b * NMAX + lane - 1] : 0.f)) : 0.f;
    float sr1 = (lane + 32 < c) ? (vs[b * NMAX + lane + 32] - vs[b * NMAX + lane + 31]) : 0.f;
    float rwd = e0 * sr0 + e1 * sr1;
#pragma unroll
    for (int off = 16; off > 0; off >>= 1) rwd += __shfl_xor(rwd, off, 32);
    rwd /= se;
    if (lane == 0) { rew[b] = rwd; out[1 + b] = rwd; }
    __syncthreads();

    if (tid == 0) {
        float loss = 0.f;
#pragma unroll
        for (int i = 0; i < 4; ++i) {
            const float x = rew[i] - rew[i + 4];
            // -log_sigmoid(x) = softplus(-x), numerically stable
            loss += fmaxf(-x, 0.f) + log1pf(expf(-fabsf(x)));
        }
        out[0] = loss * 0.25f;
    }
}

// ---------------------------------------------------------------------------
extern "C" void kernel_launch(void* const* d_in, const int* in_sizes, int n_in,
                              void* d_out, int out_size, void* d_ws, size_t ws_size,
                              hipStream_t stream)
{
    const float* hidden = (const float*)d_in[0];
    const int*   mask   = (const int*)  d_in[1];
    const float* Wq     = (const float*)d_in[2];
    const float* bq     = (const float*)d_in[3];
    const float* Wk     = (const float*)d_in[4];
    const float* bk     = (const float*)d_in[5];
    const float* Wr     = (const float*)d_in[6];
    const float* br     = (const float*)d_in[7];
    float* out = (float*)d_out;

    char* ws = (char*)d_ws;
    int*   sent_pos = (int*)ws;   ws += 4096;                 // B*64 ints (padded)
    int*   count    = (int*)ws;   ws += 256;                  // B ints (padded)
    float* ks       = (float*)ws; ws += (size_t)B * NMAX * D * sizeof(float);
    float* qv       = (float*)ws; ws += 4096;                 // B*128 floats
    float* vs       = (float*)ws;                             // B*64 floats

    k_positions <<<B, 256, 0, stream>>>(mask, sent_pos, count);
    k_kproj_wmma<<<B, 256, 0, stream>>>(hidden, Wk, bk, sent_pos, ks);
    k_small_proj<<<B, 256, 0, stream>>>(hidden, Wq, bq, Wr, br, sent_pos, qv, vs);
    k_finalize  <<<1, 256, 0, stream>>>(qv, ks, vs, count, out);
}